// Encoder_9998683865329
// MI455X (gfx1250) — compile-verified
//
#include <hip/hip_runtime.h>
#include <hip/hip_bf16.h>

typedef __attribute__((ext_vector_type(2))) float v2f;
typedef __attribute__((ext_vector_type(8))) float v8f;

#define IN_DIM  128
#define HID_DIM 64
#define OUT_DIM 128

// ---------------------------------------------------------------------------
// Degree count: deg[dst] += 1 per edge (self-loop added later as +1).
// ---------------------------------------------------------------------------
__global__ void deg_kernel(const int* __restrict__ dst, int E,
                           unsigned int* __restrict__ deg) {
    int i = blockIdx.x * blockDim.x + threadIdx.x;
    if (i < E) atomicAdd(&deg[dst[i]], 1u);
}

// dinv[i] = rsqrt(deg[i] + 1)   (deg includes self-loop -> always >= 1)
__global__ void dinv_kernel(const unsigned int* __restrict__ deg, int N,
                            float* __restrict__ dinv) {
    int i = blockIdx.x * blockDim.x + threadIdx.x;
    if (i < N) dinv[i] = rsqrtf((float)deg[i] + 1.0f);
}

// ---------------------------------------------------------------------------
// GEMM1: h[N,64] = x[N,128] @ Wc[128,64]  via V_WMMA_F32_16X16X4_F32.
// 256 threads = 8 waves; each wave computes a 16x64 tile (4 accumulators).
// A frag (16x4 f32): lane l<16 holds row l, K = k..k+1 ; lanes 16-31 K = k+2..k+3.
// B frag (4x16 f32): lane l holds col (l&15), same K split.
// C/D (16x16 f32): vgpr r -> M = r + 8*(lane>=16), N = lane&15.
// ---------------------------------------------------------------------------
__global__ __launch_bounds__(256) void gemm1_kernel(
    const float* __restrict__ x, const float* __restrict__ Wc,
    float* __restrict__ h, int N) {
    __shared__ float sW[IN_DIM * HID_DIM];           // 32 KB
    {   // cooperative stage of W_conv into LDS (2048 float4)
        const float4* gv = (const float4*)Wc;
        float4* sv = (float4*)sW;
        for (int i = threadIdx.x; i < (IN_DIM * HID_DIM) / 4; i += 256)
            sv[i] = gv[i];
    }
    __syncthreads();

    const int wave = threadIdx.x >> 5;
    const int lane = threadIdx.x & 31;
    const int rowBase = blockIdx.x * 128 + wave * 16;
    if (rowBase >= N) return;                        // wave-uniform guard

    const int r  = lane & 15;                        // row within tile / col within tile
    const int kg = lane >> 4;                        // K half select
    const int rowLd = min(rowBase + r, N - 1);       // clamped load row
    const float* xrow = x + (size_t)rowLd * IN_DIM;

    v8f acc[4] = {v8f{}, v8f{}, v8f{}, v8f{}};
    for (int k = 0; k < IN_DIM; k += 4) {
        const int kk = k + (kg << 1);
        float2 av = *(const float2*)(xrow + kk);
        v2f a; a.x = av.x; a.y = av.y;
#pragma unroll
        for (int t = 0; t < 4; ++t) {
            v2f b;
            b.x = sW[(kk)     * HID_DIM + t * 16 + r];
            b.y = sW[(kk + 1) * HID_DIM + t * 16 + r];
            acc[t] = __builtin_amdgcn_wmma_f32_16x16x4_f32(
                false, a, false, b, (short)0, acc[t], false, false);
        }
    }
#pragma unroll
    for (int t = 0; t < 4; ++t) {
        const int col = t * 16 + r;
#pragma unroll
        for (int rr = 0; rr < 8; ++rr) {
            const int row = rowBase + rr + 8 * kg;
            if (row < N) h[(size_t)row * HID_DIM + col] = acc[t][rr];
        }
    }
}

// ---------------------------------------------------------------------------
// Edge scatter: one wave per edge; lane owns 2 consecutive channels.
// agg[dst] += h[src] * dinv[src]*dinv[dst]   (h, agg are L2-resident: 2x25.6MB)
// ---------------------------------------------------------------------------
__global__ __launch_bounds__(256) void scatter_kernel(
    const int* __restrict__ src, const int* __restrict__ dst,
    const float* __restrict__ dinv, const float* __restrict__ h,
    float* __restrict__ agg, int E) {
    const int e    = (int)((blockIdx.x * (unsigned)blockDim.x + threadIdx.x) >> 5);
    const int lane = threadIdx.x & 31;
    if (e >= E) return;
    const int s = src[e];
    const int d = dst[e];
    const float nrm = dinv[s] * dinv[d];
    const float2 hv = *(const float2*)(h + (size_t)s * HID_DIM + lane * 2);
    float* ap = agg + (size_t)d * HID_DIM + lane * 2;
    atomicAdd(ap,     hv.x * nrm);
    atomicAdd(ap + 1, hv.y * nrm);
}

// ---------------------------------------------------------------------------
// GEMM2 (fused epilogue):
//   out[N,128] = relu(agg + h*dinv^2 + b_conv) @ Wl[64,128] + b_lin
// Self-loop message (h*dinv^2) and bias/relu are built into the A fragment.
// Each wave: 16x128 tile -> 8 f32 accumulators, K=64 stepped by 4.
// ---------------------------------------------------------------------------
__global__ __launch_bounds__(256) void gemm2_kernel(
    const float* __restrict__ agg, const float* __restrict__ h,
    const float* __restrict__ dinv, const float* __restrict__ bc,
    const float* __restrict__ Wl,  const float* __restrict__ bl,
    float* __restrict__ out, int N) {
    __shared__ float sW[HID_DIM * OUT_DIM];          // 32 KB
    {
        const float4* gv = (const float4*)Wl;
        float4* sv = (float4*)sW;
        for (int i = threadIdx.x; i < (HID_DIM * OUT_DIM) / 4; i += 256)
            sv[i] = gv[i];
    }
    __syncthreads();

    const int wave = threadIdx.x >> 5;
    const int lane = threadIdx.x & 31;
    const int rowBase = blockIdx.x * 128 + wave * 16;
    if (rowBase >= N) return;                        // wave-uniform guard

    const int r  = lane & 15;
    const int kg = lane >> 4;
    const int rowLd = min(rowBase + r, N - 1);
    const float* arow = agg + (size_t)rowLd * HID_DIM;
    const float* hrow = h   + (size_t)rowLd * HID_DIM;
    const float  di   = dinv[rowLd];
    const float  di2  = di * di;

    v8f acc[8] = {v8f{}, v8f{}, v8f{}, v8f{}, v8f{}, v8f{}, v8f{}, v8f{}};
    for (int k = 0; k < HID_DIM; k += 4) {
        const int kk = k + (kg << 1);
        float2 av  = *(const float2*)(arow + kk);
        float2 hv  = *(const float2*)(hrow + kk);
        float2 bcv = *(const float2*)(bc + kk);
        v2f a;
        a.x = fmaxf(av.x + hv.x * di2 + bcv.x, 0.0f);
        a.y = fmaxf(av.y + hv.y * di2 + bcv.y, 0.0f);
#pragma unroll
        for (int t = 0; t < 8; ++t) {
            v2f b;
            b.x = sW[(kk)     * OUT_DIM + t * 16 + r];
            b.y = sW[(kk + 1) * OUT_DIM + t * 16 + r];
            acc[t] = __builtin_amdgcn_wmma_f32_16x16x4_f32(
                false, a, false, b, (short)0, acc[t], false, false);
        }
    }
#pragma unroll
    for (int t = 0; t < 8; ++t) {
        const int col = t * 16 + r;
        const float blv = bl[col];
#pragma unroll
        for (int rr = 0; rr < 8; ++rr) {
            const int row = rowBase + rr + 8 * kg;
            if (row < N) out[(size_t)row * OUT_DIM + col] = acc[t][rr] + blv;
        }
    }
}

static inline size_t alignUp(size_t v) { return (v + 255) & ~(size_t)255; }

extern "C" void kernel_launch(void* const* d_in, const int* in_sizes, int n_in,
                              void* d_out, int out_size, void* d_ws, size_t ws_size,
                              hipStream_t stream) {
    const float* x  = (const float*)d_in[0];
    const int*   ei = (const int*)  d_in[1];
    const float* Wc = (const float*)d_in[2];
    const float* bc = (const float*)d_in[3];
    const float* Wl = (const float*)d_in[4];
    const float* bl = (const float*)d_in[5];
    float* out = (float*)d_out;

    const int N = in_sizes[0] / IN_DIM;     // 100000
    const int E = in_sizes[1] / 2;          // 1600000
    const int* src = ei;
    const int* dst = ei + E;

    // Workspace layout: h | agg | dinv | deg
    char* ws = (char*)d_ws;
    const size_t hBytes = (size_t)N * HID_DIM * sizeof(float);
    size_t off = 0;
    float*        h    = (float*)(ws + off);        off = alignUp(off + hBytes);
    float*        agg  = (float*)(ws + off);        off = alignUp(off + hBytes);
    float*        dinv = (float*)(ws + off);        off = alignUp(off + (size_t)N * sizeof(float));
    unsigned int* deg  = (unsigned int*)(ws + off); off = alignUp(off + (size_t)N * sizeof(unsigned int));

    hipMemsetAsync(agg, 0, hBytes, stream);
    hipMemsetAsync(deg, 0, (size_t)N * sizeof(unsigned int), stream);

    deg_kernel <<<(E + 255) / 256, 256, 0, stream>>>(dst, E, deg);
    dinv_kernel<<<(N + 255) / 256, 256, 0, stream>>>(deg, N, dinv);

    const int rowBlocks = (N + 127) / 128;          // 8 waves x 16 rows per block
    gemm1_kernel<<<rowBlocks, 256, 0, stream>>>(x, Wc, h, N);

    scatter_kernel<<<(E + 7) / 8, 256, 0, stream>>>(src, dst, dinv, h, agg, E);

    gemm2_kernel<<<rowBlocks, 256, 0, stream>>>(agg, h, dinv, bc, Wl, bl, out, N);
}